// DeepRNNC_86535001080504
// MI455X (gfx1250) — compile-verified
//
#include <hip/hip_runtime.h>
#include <hip/hip_bf16.h>

typedef __attribute__((ext_vector_type(2))) float v2f;
typedef __attribute__((ext_vector_type(8))) float v8f;

#define H20   20
#define BATCH 512
#define TLEN  2048
#define NOUT  10
#define WPB   4            // waves per block
#define HB    (16 * H20 + 32)   // per-wave buffer: 16x20 matrix + 32-float dump region
#define DUMP  (16 * H20)

__device__ __forceinline__ float fast_tanh(float x) {
#if __has_builtin(__builtin_amdgcn_tanhf)
    return __builtin_amdgcn_tanhf(x);          // gfx1250 v_tanh_f32 (single trans op)
#elif __has_builtin(__builtin_amdgcn_rcpf)
    float e = __expf(2.0f * x);
    return 1.0f - 2.0f * __builtin_amdgcn_rcpf(e + 1.0f);
#else
    float e = __expf(2.0f * x);
    return 1.0f - 2.0f / (e + 1.0f);
#endif
}

__device__ __forceinline__ v8f wmma_f32(v2f a, v2f b, v8f c) {
    // V_WMMA_F32_16X16X4_F32 : D(16x16 f32) = A(16x4 f32) x B(4x16 f32) + C
    return __builtin_amdgcn_wmma_f32_16x16x4_f32(
        false, a, false, b, (short)0, c, false, false);
}

__device__ __forceinline__ v8f vadd8(v8f a, v8f b) {
    v8f r;
    #pragma unroll
    for (int i = 0; i < 8; ++i) r[i] = a[i] + b[i];        // -> v_pk_add_f32 x4
    return r;
}

__global__ __launch_bounds__(WPB * 32)
void deep_rnn_fused(const float* __restrict__ x,
                    const float* __restrict__ W_ih0,
                    const float* __restrict__ W_hh0,
                    const float* __restrict__ b_ih0,
                    const float* __restrict__ b_hh0,
                    const float* __restrict__ W_ih1,
                    const float* __restrict__ W_hh1,
                    const float* __restrict__ b_ih1,
                    const float* __restrict__ b_hh1,
                    const float* __restrict__ W_fc,
                    const float* __restrict__ b_fc,
                    float* __restrict__ out)
{
    __shared__ float sh[WPB][2][HB];         // per-wave h0 / h1 transpose buffers + dump

    const int lane  = threadIdx.x & 31;
    const int wave  = threadIdx.x >> 5;
    const int gwave = blockIdx.x * WPB + wave;
    const int bB    = gwave * 16;            // batch tile base
    const int m     = lane & 15;             // A-row / B,D-col within tile
    const int half  = lane >> 4;
    const int hi8   = half * 8;              // D: M offset for lanes 16-31
    const int j0    = half * 2;              // K pair base within a 4-chunk
    const float sel = (half == 0) ? 1.0f : 0.0f;   // aug chunk lives only in K rows 20,21

    float* h0buf = &sh[wave][0][0];
    float* h1buf = &sh[wave][1][0];

    // per-lane store indices (constant over the whole T loop).
    // tile0: D VGPR r -> h[r+hi8][m]; tile1: -> h[r+hi8][16+m] for m<4, else dump slot.
    int st0[8], st1[8];
    #pragma unroll
    for (int r = 0; r < 8; ++r) {
        st0[r] = (r + hi8) * H20 + m;
        st1[r] = (m < 4) ? ((r + hi8) * H20 + 16 + m) : (DUMP + lane);
    }

    // ------- hoisted weight B-fragments.  B[k][n]: lanes hold N, VGPR pair holds K -------
    // K augmented to 24: rows 0..19 = W^T, rows 20/21 = fused input-weight / bias, 22..23 = 0
    v2f wbhh0[6][2], wbhh1[6][2], wbih1[5][2], wbfc[6];
    #pragma unroll
    for (int nt = 0; nt < 2; ++nt) {
        const int  g  = nt * 16 + m;
        const bool gv = (g < H20);
        #pragma unroll
        for (int kc = 0; kc < 5; ++kc) {
            const int j = kc * 4 + j0;             // j, j+1 <= 19 always
            v2f w;
            w.x = gv ? W_hh0[g * H20 + j]     : 0.0f;
            w.y = gv ? W_hh0[g * H20 + j + 1] : 0.0f;
            wbhh0[kc][nt] = w;
            w.x = gv ? W_ih1[g * H20 + j]     : 0.0f;
            w.y = gv ? W_ih1[g * H20 + j + 1] : 0.0f;
            wbih1[kc][nt] = w;
            w.x = gv ? W_hh1[g * H20 + j]     : 0.0f;
            w.y = gv ? W_hh1[g * H20 + j + 1] : 0.0f;
            wbhh1[kc][nt] = w;
        }
        v2f w;
        w.x = (gv && half == 0) ? W_ih0[g] : 0.0f;                    // row 20: input weight
        w.y = (gv && half == 0) ? (b_ih0[g] + b_hh0[g]) : 0.0f;       // row 21: fused bias
        wbhh0[5][nt] = w;
        w.x = (gv && half == 0) ? (b_ih1[g] + b_hh1[g]) : 0.0f;       // row 20: layer1 bias
        w.y = 0.0f;
        wbhh1[5][nt] = w;
    }
    {
        const bool ov = (m < NOUT);
        #pragma unroll
        for (int kc = 0; kc < 5; ++kc) {
            const int j = kc * 4 + j0;
            v2f w;
            w.x = ov ? W_fc[m * H20 + j]     : 0.0f;
            w.y = ov ? W_fc[m * H20 + j + 1] : 0.0f;
            wbfc[kc] = w;
        }
        v2f w;
        w.x = (ov && half == 0) ? b_fc[m] : 0.0f;                     // row 20: fc bias
        w.y = 0.0f;
        wbfc[5] = w;
    }

    // A-fragments.  chunk 5 of a0 = {x*sel, sel}; chunk 5 of a1 = {sel, 0} (constant)
    v2f a0[6], a1[6];
    #pragma unroll
    for (int kc = 0; kc < 6; ++kc) {
        v2f z; z.x = 0.0f; z.y = 0.0f;
        a0[kc] = z; a1[kc] = z;
    }
    a1[5].x = sel;
    const v8f vzero = {};

    const float* xp    = x + (size_t)(bB + m) * TLEN;     // lane m -> batch bB+m
    float*       obase = out + (size_t)bB * TLEN * NOUT;

    for (int t = 0; t < TLEN; ++t, ++xp) {
        // ============ layer 0: h0 = tanh([h0; x; 1] @ Whh0_aug^T), C starts inline-0 ============
        const float xl = *xp;
        a0[5].x = xl * sel;
        a0[5].y = sel;

        // two independent 3-deep chains per N-tile (4 streams), then merge
        v8f pA0 = vzero, pB0 = vzero, pA1 = vzero, pB1 = vzero;
        #pragma unroll
        for (int kc = 0; kc < 3; ++kc) {
            pA0 = wmma_f32(a0[kc],     wbhh0[kc][0],     pA0);
            pA1 = wmma_f32(a0[kc],     wbhh0[kc][1],     pA1);
            pB0 = wmma_f32(a0[kc + 3], wbhh0[kc + 3][0], pB0);
            pB1 = wmma_f32(a0[kc + 3], wbhh0[kc + 3][1], pB1);
        }
        v8f acc0 = vadd8(pA0, pB0);
        v8f acc1 = vadd8(pA1, pB1);
        #pragma unroll
        for (int r = 0; r < 8; ++r) { acc0[r] = fast_tanh(acc0[r]); acc1[r] = fast_tanh(acc1[r]); }

        // D -> row-major [16][20] in LDS (unconditional; invalid lanes hit dump slots)
        #pragma unroll
        for (int r = 0; r < 8; ++r) h0buf[st0[r]] = acc0[r];
        #pragma unroll
        for (int r = 0; r < 8; ++r) h0buf[st1[r]] = acc1[r];

        // h1 @ Whh1_aug chains use LAST step's a1 -> issue while the a0 reload is in flight
        v8f cA0 = vzero, cA1 = vzero;
        #pragma unroll
        for (int kc = 0; kc < 6; ++kc) {
            cA0 = wmma_f32(a1[kc], wbhh1[kc][0], cA0);
            cA1 = wmma_f32(a1[kc], wbhh1[kc][1], cA1);
        }

        #pragma unroll
        for (int kc = 0; kc < 5; ++kc)
            a0[kc] = *(const v2f*)&h0buf[m * H20 + kc * 4 + j0];   // ds_load_b64

        // ============ layer 1: h1 = tanh(h0 @ Wih1^T + [h1;1] @ Whh1_aug^T) ============
        // fresh-a0 side split 3+2 so the post-LDS dependent chain is only 3 deep
        v8f cB0 = vzero, cB1 = vzero, cC0 = vzero, cC1 = vzero;
        #pragma unroll
        for (int kc = 0; kc < 3; ++kc) {
            cB0 = wmma_f32(a0[kc], wbih1[kc][0], cB0);
            cB1 = wmma_f32(a0[kc], wbih1[kc][1], cB1);
        }
        #pragma unroll
        for (int kc = 3; kc < 5; ++kc) {
            cC0 = wmma_f32(a0[kc], wbih1[kc][0], cC0);
            cC1 = wmma_f32(a0[kc], wbih1[kc][1], cC1);
        }
        v8f c0 = vadd8(vadd8(cA0, cB0), cC0);
        v8f c1 = vadd8(vadd8(cA1, cB1), cC1);
        #pragma unroll
        for (int r = 0; r < 8; ++r) { c0[r] = fast_tanh(c0[r]); c1[r] = fast_tanh(c1[r]); }

        #pragma unroll
        for (int r = 0; r < 8; ++r) h1buf[st0[r]] = c0[r];
        #pragma unroll
        for (int r = 0; r < 8; ++r) h1buf[st1[r]] = c1[r];
        #pragma unroll
        for (int kc = 0; kc < 5; ++kc)
            a1[kc] = *(const v2f*)&h1buf[m * H20 + kc * 4 + j0];

        // ============ head: out[b,t,:] = [h1;1] @ Wfc_aug^T (bias rides aug row) ============
        v8f oacc = vzero;
        #pragma unroll
        for (int kc = 0; kc < 6; ++kc)
            oacc = wmma_f32(a1[kc], wbfc[kc], oacc);

        if (m < NOUT) {
            #pragma unroll
            for (int r = 0; r < 8; ++r)
                obase[(size_t)(r + hi8) * TLEN * NOUT + (size_t)t * NOUT + m] = oacc[r];
        }
    }
}

extern "C" void kernel_launch(void* const* d_in, const int* in_sizes, int n_in,
                              void* d_out, int out_size, void* d_ws, size_t ws_size,
                              hipStream_t stream) {
    (void)in_sizes; (void)n_in; (void)out_size; (void)d_ws; (void)ws_size;
    const float* x     = (const float*)d_in[0];
    const float* W_ih0 = (const float*)d_in[1];
    const float* W_hh0 = (const float*)d_in[2];
    const float* b_ih0 = (const float*)d_in[3];
    const float* b_hh0 = (const float*)d_in[4];
    const float* W_ih1 = (const float*)d_in[5];
    const float* W_hh1 = (const float*)d_in[6];
    const float* b_ih1 = (const float*)d_in[7];
    const float* b_hh1 = (const float*)d_in[8];
    const float* W_fc  = (const float*)d_in[9];
    const float* b_fc  = (const float*)d_in[10];
    float* out = (float*)d_out;

    const int waves  = BATCH / 16;           // 32 batch tiles of 16
    const int blocks = waves / WPB;          // 8 blocks x 128 threads
    deep_rnn_fused<<<blocks, WPB * 32, 0, stream>>>(
        x, W_ih0, W_hh0, b_ih0, b_hh0,
        W_ih1, W_hh1, b_ih1, b_hh1, W_fc, b_fc, out);
}